// MonarchLinear_17042430230557
// MI455X (gfx1250) — compile-verified
//
#include <hip/hip_runtime.h>

// MI455X / gfx1250 Monarch linear, fp32 WMMA (16x16x4) fused two-stage kernel.
//
//   B=16384, IN=256, OUT=4608, M=68, PAD=4624
//   stage1: t1[b,r,l] = sum_p x[b,68r+p] * L[r,l,p]   (only r=0..3 nonzero)
//   stage2: y[b, s*68+l] = sum_{r<4} t1[b,r,l] * R[l,s,r] + bias[s*68+l]
//
// Roofline: ~1.2 GFLOP structured compute vs 16 MB x-read + 302 MB y-write
// -> pure store-bandwidth bound (~14 us at 23.3 TB/s). Weights (2.5 MB) are
// L2-resident; y is streamed with non-temporal stores.

typedef float v2f __attribute__((ext_vector_type(2)));
typedef float v4f __attribute__((ext_vector_type(4)));
typedef float v8f __attribute__((ext_vector_type(8)));

#define NB     16384
#define IND    256
#define OUTD   4608
#define MDIM   68
#define RPAD   4624        // 68*68

// LDS strides (dwords), padded for bank spread
#define AS_ROW 276         // >= 68*4 = 272
#define YS_ROW 320         // >= 16*20
#define YS_S   20          // >= 16 (l slots), multiple of 4 for b128 alignment

#define WMMA_F32(a, b, c) \
    __builtin_amdgcn_wmma_f32_16x16x4_f32(false, (a), false, (b), (short)0, (c), false, false)

// ---------------- Stage 2 helper: one l-group of width LW ----------------
// LW=16/QN=4 for the four full groups, LW=4/QN=1 for the tail (lg=64).
template <int LW, int QN>
__device__ __forceinline__ void stage2_group(const float* __restrict__ as,
                                             float* __restrict__ ys,
                                             const float* __restrict__ R,
                                             const float* __restrict__ bias,
                                             float* __restrict__ y,
                                             int lg, int rowbase,
                                             int lane, int half, int ln)
{
    for (int st = 0; st < 5; ++st) {
        const int sbase = st * 16;
        const int se = (sbase + ln > MDIM - 1) ? (MDIM - 1) : (sbase + ln); // clamped gather lane

#pragma unroll 4
        for (int li = 0; li < LW; ++li) {
            const int l = lg + li;
            // A frag: M=ln, K=r = 2*half + {0,1} from As[row=ln][l][r]
            v2f a = *(const v2f*)(as + ln * AS_ROW + l * 4 + 2 * half);
            // B frag: N=ln, K=r from R[l, sbase+ln, r]
            v2f b = *(const v2f*)(R + (size_t)l * RPAD + (size_t)se * MDIM + 2 * half);
            // C init = bias broadcast along M
            int o = se * MDIM + l;
            if (o > OUTD - 1) o = OUTD - 1;
            const float bb = bias[o];
            v8f c = { bb, bb, bb, bb, bb, bb, bb, bb };
            c = WMMA_F32(a, b, c);
            // D -> Ys[row][s=ln][li]
#pragma unroll
            for (int v = 0; v < 8; ++v)
                ys[(v + 8 * half) * YS_ROW + ln * YS_S + li] = c[v];
        }

        // Coalesced store phase: 16B chunks of 4 consecutive l (non-temporal stream).
        const int chunks = 16 * 16 * QN;
        for (int t = lane; t < chunks; t += 32) {
            const int q   = t % QN;              // QN is 4 or 1 -> shifts
            const int s   = (t / QN) & 15;
            const int row = t / (QN * 16);
            const int sg  = sbase + s;
            const int l0  = lg + 4 * q;
            // output cut o<4608 is chunk-aligned (s=67 admits l<=51)
            if (sg < MDIM && (sg * MDIM + l0 + 3) < OUTD) {
                v4f vv = *(const v4f*)(ys + row * YS_ROW + s * YS_S + 4 * q);
                __builtin_nontemporal_store(
                    vv, (v4f*)(y + (size_t)(rowbase + row) * OUTD + sg * MDIM + l0));
            }
        }
    }
}

__global__ __launch_bounds__(128, 1)
void monarch_wmma_kernel(const float* __restrict__ x,
                         const float* __restrict__ L,
                         const float* __restrict__ R,
                         const float* __restrict__ bias,
                         float* __restrict__ y)
{
    __shared__ float As[4][16 * AS_ROW];   // per-wave: t1 as [row16][l68][r4]
    __shared__ float Ys[4][16 * YS_ROW];   // per-wave: [row16][s16][l16] transpose buf

    const int lane = threadIdx.x & 31;
    const int wid  = threadIdx.x >> 5;
    const int half = lane >> 4;            // 0: lanes 0-15, 1: lanes 16-31
    const int ln   = lane & 15;
    const int rowbase = blockIdx.x * 64 + wid * 16;

    float* as = As[wid];
    float* ys = Ys[wid];

    // ---------------- Stage 1: t1[b,r,l] via WMMA f32 16x16x4 ----------------
    // ks outermost per r: ONE register-resident A fragment feeds 5 l-tile
    // accumulators -> no fragment array, no scratch spills.
#pragma unroll
    for (int r = 0; r < 4; ++r) {
        const int ksteps = (r < 3) ? 17 : 13;   // r=3 block: only p=0..51 maps into x[0..255]
        v8f c0 = {}, c1 = {}, c2 = {}, c3 = {}, c4 = {};
        const float* xp = x + (size_t)(rowbase + ln) * IND + r * MDIM + 2 * half;
        const float* lp = L + (size_t)(r * MDIM + ln) * MDIM + 2 * half;

#pragma unroll
        for (int ks = 0; ks < ksteps; ++ks) {
            // A frag: lane(half,ln): M=ln, K=2*half+{0,1} -> x[row, 68r+4ks+2*half+{0,1}]
            v2f a  = *(const v2f*)(xp + 4 * ks);
            // B frags: lane: N=ln, K=2*half+{0,1} -> L[r, lt*16+ln, 4ks+2*half+{0,1}]
            v2f b0 = *(const v2f*)(lp + 4 * ks);
            v2f b1 = *(const v2f*)(lp + 1 * 16 * MDIM + 4 * ks);
            v2f b2 = *(const v2f*)(lp + 2 * 16 * MDIM + 4 * ks);
            v2f b3 = *(const v2f*)(lp + 3 * 16 * MDIM + 4 * ks);
            v2f b4 = *(const v2f*)(lp + 4 * 16 * MDIM + 4 * ks);
            c0 = WMMA_F32(a, b0, c0);
            c1 = WMMA_F32(a, b1, c1);
            c2 = WMMA_F32(a, b2, c2);
            c3 = WMMA_F32(a, b3, c3);
            c4 = WMMA_F32(a, b4, c4);
        }

        // D: VGPR v -> row = v + 8*half, col l = lt*16 + ln; store As[row][l][r]
#define STORE_C(CREG, LT)                                                    \
        do {                                                                 \
            const int l_ = (LT) * 16 + ln;                                   \
            if (l_ < MDIM) {                                                 \
                _Pragma("unroll")                                            \
                for (int v = 0; v < 8; ++v)                                  \
                    as[(v + 8 * half) * AS_ROW + l_ * 4 + r] = (CREG)[v];    \
            }                                                                \
        } while (0)
        STORE_C(c0, 0); STORE_C(c1, 1); STORE_C(c2, 2); STORE_C(c3, 3); STORE_C(c4, 4);
#undef STORE_C
    }
    // wave-private LDS: ds ops in-order within the wave; no barrier needed.

    // ---------------- Stage 2: one WMMA (K=4) per (l, s-tile) ----------------
    for (int lg = 0; lg < 64; lg += 16)
        stage2_group<16, 4>(as, ys, R, bias, y, lg, rowbase, lane, half, ln);
    stage2_group<4, 1>(as, ys, R, bias, y, 64, rowbase, lane, half, ln);
}

extern "C" void kernel_launch(void* const* d_in, const int* in_sizes, int n_in,
                              void* d_out, int out_size, void* d_ws, size_t ws_size,
                              hipStream_t stream) {
    (void)in_sizes; (void)n_in; (void)d_ws; (void)ws_size; (void)out_size;
    const float* x    = (const float*)d_in[0];
    const float* L    = (const float*)d_in[1];
    const float* R    = (const float*)d_in[2];
    const float* bias = (const float*)d_in[3];
    float* y = (float*)d_out;

    dim3 grid(NB / 64);   // 256 blocks, 64 batch rows each
    dim3 block(128);      // 4 wave32, 16 rows per wave
    hipLaunchKernelGGL(monarch_wmma_kernel, grid, block, 0, stream, x, L, R, bias, y);
}